// LSTMDecoder_34033320854143
// MI455X (gfx1250) — compile-verified
//
#include <hip/hip_runtime.h>

// LSTM decoder with Bahdanau attention for MI455X (gfx1250, wave32).
// GEMMs: V_WMMA_F32_16X16X4_F32 (fp32 matrix path) fed from LDS slabs that
// are staged with global_load_async_to_lds_b128 (ASYNCcnt) + double buffering.
// Inner loops batch all fragment loads ahead of the WMMA group so the LDS/VMEM
// pipe overlaps the matrix pipe instead of ping-ponging on s_wait_dscnt.
//
// Dimensions (compile-time constants from the reference):
#define B_ 16
#define T_ 128
#define S_ 128
#define E_ 512
#define H_ 512
#define V_ 4096

typedef __attribute__((ext_vector_type(2))) float v2f;
typedef __attribute__((ext_vector_type(8))) float v8f;

// D = A(16x4) * B(4x16) + C(16x16), fp32.
// (neg_a, A, neg_b, B, c_mod, C, reuse_a, reuse_b)
__device__ __forceinline__ v8f wmma4(v2f a, v2f b, v8f c) {
  return __builtin_amdgcn_wmma_f32_16x16x4_f32(false, a, false, b, (short)0, c,
                                               false, false);
}

__device__ __forceinline__ float sigmoidf_(float x) {
  return 1.0f / (1.0f + __expf(-x));
}

// Async copy global -> LDS, 16 bytes per lane. VDST = per-lane LDS byte
// address, VADDR = per-lane 64-bit global address (GV mode). ASYNCcnt.
__device__ __forceinline__ void async_ld_b128(unsigned lds_byte_addr,
                                              const float* g) {
  asm volatile("global_load_async_to_lds_b128 %0, %1, off" ::"v"(lds_byte_addr),
               "v"((unsigned long long)(uintptr_t)g)
               : "memory");
}
__device__ __forceinline__ void wait_async0() {
  asm volatile("s_wait_asynccnt 0x0" ::: "memory");
}

// Fragment layouts (ISA 7.12.2, wave32):
//   A 16x4 : lane L<16 -> row M=L, v.x=K0 v.y=K1 ; lane L>=16 -> K2/K3
//   B 4x16 : lane L<16 -> col N=L, v.x=K0 v.y=K1 ; lane L>=16 -> K2/K3
//   C 16x16: element r of v8f -> M = r + 8*(lane>=16), N = lane&15

// ---------------------------------------------------------------------------
// x[bt,:] = emb[tgt[bt],:]   (grid = B*T blocks, block = E/4 threads, float4)
__global__ void embed_kernel(const int* __restrict__ tgt,
                             const float* __restrict__ emb,
                             float* __restrict__ x) {
  int bt = blockIdx.x;
  int tok = tgt[bt];
  const float4* s = (const float4*)(emb + (size_t)tok * E_);
  float4* d = (float4*)(x + (size_t)bt * E_);
  d[threadIdx.x] = s[threadIdx.x];
}

// enc_hidden = mean_s memory[b,s,:] ; c0 = c1 = 0
__global__ void init_kernel(const float* __restrict__ memory,
                            float* __restrict__ hinit, float* __restrict__ c0,
                            float* __restrict__ c1) {
  int idx = blockIdx.x * blockDim.x + threadIdx.x;  // 0 .. B*H-1
  int b = idx >> 9;
  int e = idx & (H_ - 1);
  const float* mp = memory + (size_t)b * S_ * E_ + e;
  float s = 0.f;
  for (int ss = 0; ss < S_; ++ss) s += mp[(size_t)ss * E_];
  hinit[idx] = s * (1.0f / (float)S_);
  c0[idx] = 0.f;
  c1[idx] = 0.f;
}

// ---------------------------------------------------------------------------
// LDS-staged block GEMM: C = [A1|A2] @ W^T + bias1 + bias2.
//   A1: [M,K1] lda1, A2: [M,K2] lda2 (nullable), W: [N, K1+K2] ldw row-major.
// Block = 128 threads (4 waves) computes a TM x TN = 32x128 C tile.
// K-slabs of TK=32 are async-copied into double-buffered LDS (rows padded to
// PAD=36 floats -> conflict-free 64-bank fragment reads). Each wave owns one
// 16-row half and four 16-col subtiles: 4 independent WMMA chains.
#define TM 32
#define TN 128
#define TK 32
#define PAD 36
#define GEMM_LDS_BYTES ((2 * TM * PAD + 2 * TN * PAD) * 4)  // 46080 B

__global__ void gemm_wt_kernel(const float* __restrict__ A1, int lda1, int K1,
                               const float* __restrict__ A2, int lda2, int K2,
                               const float* __restrict__ W, int ldw,
                               const float* __restrict__ bias1,
                               const float* __restrict__ bias2,
                               float* __restrict__ C, int ldc) {
  extern __shared__ float smem[];
  // Generic LDS pointer low 32 bits == LDS byte offset (flat->LDS aperture).
  const unsigned lds_base = (unsigned)(uintptr_t)smem;

  const int tid = threadIdx.x;  // 0..127
  const int lane = tid & 31;
  const int wave = tid >> 5;  // 0..3
  const int m0 = blockIdx.x * TM;
  const int n0 = blockIdx.y * TN;
  const int mr = lane & 15;
  const int kh = (lane >> 4) << 1;
  const int mo = (lane >> 4) << 3;
  const int msub = wave & 1;          // which 16-row half of the A tile
  const int nbase = (wave >> 1) * 4;  // first of 4 owned 16-col subtiles

  const int nslabs = (K1 + K2) / TK;
  const int r0 = tid >> 3;        // 0..15: load row group
  const int c4 = (tid & 7) << 2;  // float col within slab: 0,4,..,28

  // LDS float offsets of the four buffers.
  const int A_buf[2] = {0, TM * PAD};
  const int B_buf[2] = {2 * TM * PAD, 2 * TM * PAD + TN * PAD};

  v8f acc[4];
#pragma unroll
  for (int t = 0; t < 4; ++t)
    acc[t] = (v8f){0.f, 0.f, 0.f, 0.f, 0.f, 0.f, 0.f, 0.f};

  auto issue_slab = [&](int s, int buf) {
    const int kb = s * TK;
    const float* Abase;
    int lda, kcol;
    if (kb < K1) { Abase = A1; lda = lda1; kcol = kb; }
    else         { Abase = A2; lda = lda2; kcol = kb - K1; }
#pragma unroll
    for (int i = 0; i < 2; ++i) {  // A slab: 32 rows x 32 floats
      int row = r0 + 16 * i;
      const float* g = Abase + (size_t)(m0 + row) * lda + kcol + c4;
      async_ld_b128(lds_base + (unsigned)((A_buf[buf] + row * PAD + c4) * 4), g);
    }
#pragma unroll
    for (int i = 0; i < 8; ++i) {  // W slab: 128 rows x 32 floats
      int row = r0 + 16 * i;
      const float* g = W + (size_t)(n0 + row) * ldw + kb + c4;
      async_ld_b128(lds_base + (unsigned)((B_buf[buf] + row * PAD + c4) * 4), g);
    }
  };

  issue_slab(0, 0);
  for (int s = 0; s < nslabs; ++s) {
    wait_async0();    // this wave's slab-s data is in LDS
    __syncthreads();  // everyone's slab-s data in; prior buffer reads done
    if (s + 1 < nslabs) issue_slab(s + 1, (s + 1) & 1);
    const float* sA = smem + A_buf[s & 1] + (msub * 16 + mr) * PAD + kh;
    const float* sB = smem + B_buf[s & 1] + (nbase * 16 + mr) * PAD + kh;
#pragma unroll
    for (int k4 = 0; k4 < TK; k4 += 4) {
      // Batch all 5 fragment loads, then fire 4 independent WMMAs.
      v2f a, b0, b1, b2, b3;
      a.x = sA[k4];                 a.y = sA[k4 + 1];
      b0.x = sB[k4];                b0.y = sB[k4 + 1];
      b1.x = sB[16 * PAD + k4];     b1.y = sB[16 * PAD + k4 + 1];
      b2.x = sB[32 * PAD + k4];     b2.y = sB[32 * PAD + k4 + 1];
      b3.x = sB[48 * PAD + k4];     b3.y = sB[48 * PAD + k4 + 1];
      acc[0] = wmma4(a, b0, acc[0]);
      acc[1] = wmma4(a, b1, acc[1]);
      acc[2] = wmma4(a, b2, acc[2]);
      acc[3] = wmma4(a, b3, acc[3]);
    }
  }

  const int mbase = m0 + msub * 16;
#pragma unroll
  for (int t = 0; t < 4; ++t) {
    int nc = n0 + (nbase + t) * 16 + mr;
    float bb = (bias1 ? bias1[nc] : 0.f) + (bias2 ? bias2[nc] : 0.f);
#pragma unroll
    for (int r = 0; r < 8; ++r)
      C[(size_t)(mbase + r + mo) * ldc + nc] = acc[t][r] + bb;
  }
}

// ---------------------------------------------------------------------------
// LSTM layer-0 step: gates = XG(t) + h_prev @ Whh^T, then elementwise update.
// grid = H/16 workgroups of 32 threads; each owns a 16-wide hidden slice and
// the 4 corresponding gate tiles (i,f,g,o) -> 4 independent WMMA chains.
// c is updated in place (slice-private); h is written to y[:,t,:].
__global__ void lstm_step0_kernel(const float* __restrict__ xg, long xg_stride,
                                  const float* __restrict__ hprev, long h_stride,
                                  const float* __restrict__ Whh,
                                  float* __restrict__ cbuf,
                                  float* __restrict__ hout, long hout_stride) {
  const int lane = threadIdx.x;
  const int j0 = blockIdx.x * 16;
  const int mr = lane & 15;
  const int kh = (lane >> 4) << 1;
  const int mo = (lane >> 4) << 3;
  v8f acc[4];
#pragma unroll
  for (int g = 0; g < 4; ++g) {
    const int n0 = g * H_ + j0;
#pragma unroll
    for (int r = 0; r < 8; ++r)
      acc[g][r] = xg[(size_t)(r + mo) * xg_stride + n0 + mr];
  }
  const float* hr = hprev + (size_t)mr * h_stride + kh;
  const float* w0 = Whh + (size_t)(0 * H_ + j0 + mr) * H_ + kh;
  const float* w1 = Whh + (size_t)(1 * H_ + j0 + mr) * H_ + kh;
  const float* w2 = Whh + (size_t)(2 * H_ + j0 + mr) * H_ + kh;
  const float* w3 = Whh + (size_t)(3 * H_ + j0 + mr) * H_ + kh;
  for (int k = 0; k < H_; k += 4) {
    v2f a, b0, b1, b2, b3;
    a.x = hr[k];  a.y = hr[k + 1];
    b0.x = w0[k]; b0.y = w0[k + 1];
    b1.x = w1[k]; b1.y = w1[k + 1];
    b2.x = w2[k]; b2.y = w2[k + 1];
    b3.x = w3[k]; b3.y = w3[k + 1];
    acc[0] = wmma4(a, b0, acc[0]);
    acc[1] = wmma4(a, b1, acc[1]);
    acc[2] = wmma4(a, b2, acc[2]);
    acc[3] = wmma4(a, b3, acc[3]);
  }
  const int j = j0 + mr;
#pragma unroll
  for (int r = 0; r < 8; ++r) {
    int m = r + mo;
    float ig = sigmoidf_(acc[0][r]);
    float fg = sigmoidf_(acc[1][r]);
    float gg = tanhf(acc[2][r]);
    float og = sigmoidf_(acc[3][r]);
    float cn = fg * cbuf[m * H_ + j] + ig * gg;
    cbuf[m * H_ + j] = cn;
    hout[(size_t)m * hout_stride + j] = og * tanhf(cn);
  }
}

// LSTM layer-1 step: gates = x_t @ Wih^T + h_prev @ Whh^T + (bih+bhh).
__global__ void lstm_step1_kernel(const float* __restrict__ xin, long x_stride,
                                  const float* __restrict__ Wih,
                                  const float* __restrict__ hprev, long h_stride,
                                  const float* __restrict__ Whh,
                                  const float* __restrict__ bih,
                                  const float* __restrict__ bhh,
                                  float* __restrict__ cbuf,
                                  float* __restrict__ hout, long hout_stride) {
  const int lane = threadIdx.x;
  const int j0 = blockIdx.x * 16;
  const int mr = lane & 15;
  const int kh = (lane >> 4) << 1;
  const int mo = (lane >> 4) << 3;
  v8f acc[4];
#pragma unroll
  for (int g = 0; g < 4; ++g) {
    float bsum = bih[g * H_ + j0 + mr] + bhh[g * H_ + j0 + mr];
#pragma unroll
    for (int r = 0; r < 8; ++r) acc[g][r] = bsum;  // bias depends only on N
  }
  const float* xr = xin + (size_t)mr * x_stride + kh;
  const float* hr = hprev + (size_t)mr * h_stride + kh;
  const float* wi0 = Wih + (size_t)(0 * H_ + j0 + mr) * H_ + kh;
  const float* wi1 = Wih + (size_t)(1 * H_ + j0 + mr) * H_ + kh;
  const float* wi2 = Wih + (size_t)(2 * H_ + j0 + mr) * H_ + kh;
  const float* wi3 = Wih + (size_t)(3 * H_ + j0 + mr) * H_ + kh;
  const float* wh0 = Whh + (size_t)(0 * H_ + j0 + mr) * H_ + kh;
  const float* wh1 = Whh + (size_t)(1 * H_ + j0 + mr) * H_ + kh;
  const float* wh2 = Whh + (size_t)(2 * H_ + j0 + mr) * H_ + kh;
  const float* wh3 = Whh + (size_t)(3 * H_ + j0 + mr) * H_ + kh;
  for (int k = 0; k < H_; k += 4) {
    v2f ax, ah, bi0, bi1, bi2, bi3, bh0, bh1, bh2, bh3;
    ax.x = hr[k] * 0.f + xr[k];  // keep simple loads; no arithmetic fusion
    ax.x = xr[k];   ax.y = xr[k + 1];
    ah.x = hr[k];   ah.y = hr[k + 1];
    bi0.x = wi0[k]; bi0.y = wi0[k + 1];
    bi1.x = wi1[k]; bi1.y = wi1[k + 1];
    bi2.x = wi2[k]; bi2.y = wi2[k + 1];
    bi3.x = wi3[k]; bi3.y = wi3[k + 1];
    bh0.x = wh0[k]; bh0.y = wh0[k + 1];
    bh1.x = wh1[k]; bh1.y = wh1[k + 1];
    bh2.x = wh2[k]; bh2.y = wh2[k + 1];
    bh3.x = wh3[k]; bh3.y = wh3[k + 1];
    acc[0] = wmma4(ax, bi0, acc[0]);
    acc[1] = wmma4(ax, bi1, acc[1]);
    acc[2] = wmma4(ax, bi2, acc[2]);
    acc[3] = wmma4(ax, bi3, acc[3]);
    acc[0] = wmma4(ah, bh0, acc[0]);
    acc[1] = wmma4(ah, bh1, acc[1]);
    acc[2] = wmma4(ah, bh2, acc[2]);
    acc[3] = wmma4(ah, bh3, acc[3]);
  }
  const int j = j0 + mr;
#pragma unroll
  for (int r = 0; r < 8; ++r) {
    int m = r + mo;
    float ig = sigmoidf_(acc[0][r]);
    float fg = sigmoidf_(acc[1][r]);
    float gg = tanhf(acc[2][r]);
    float og = sigmoidf_(acc[3][r]);
    float cn = fg * cbuf[m * H_ + j] + ig * gg;
    cbuf[m * H_ + j] = cn;
    hout[(size_t)m * hout_stride + j] = og * tanhf(cn);
  }
}

// ---------------------------------------------------------------------------
// Fused energy + softmax + context for one (b,t). block = 256 (8 waves).
__global__ void attn_kernel(const float* __restrict__ enc_feat,
                            const float* __restrict__ dec_feat,
                            const float* __restrict__ memory,
                            const unsigned char* __restrict__ mask,
                            const float* __restrict__ v,
                            const float* __restrict__ vb,
                            float* __restrict__ context) {
  __shared__ float df[H_];
  __shared__ float en[S_];
  __shared__ float smax, ssum;
  const int bt = blockIdx.x;
  const int b = bt >> 7;  // / T_
  const int tid = threadIdx.x;
  for (int h = tid; h < H_; h += 256) df[h] = dec_feat[(size_t)bt * H_ + h];
  __syncthreads();
  const int wave = tid >> 5, lane = tid & 31;
  for (int s = wave; s < S_; s += 8) {
    const float* ef = enc_feat + ((size_t)b * S_ + s) * H_;
    float sum = 0.f;
    for (int h = lane; h < H_; h += 32) sum += tanhf(ef[h] + df[h]) * v[h];
    for (int off = 16; off > 0; off >>= 1) sum += __shfl_down(sum, off);
    if (lane == 0) {
      float e = sum + vb[0];
      if (mask[b * S_ + s]) e = -1e9f;
      en[s] = e;
    }
  }
  __syncthreads();
  if (tid == 0) {
    float mx = en[0];
    for (int s = 1; s < S_; ++s) mx = fmaxf(mx, en[s]);
    smax = mx;
  }
  __syncthreads();
  if (tid < S_) en[tid] = __expf(en[tid] - smax);
  __syncthreads();
  if (tid == 0) {
    float sm = 0.f;
    for (int s = 0; s < S_; ++s) sm += en[s];
    ssum = 1.0f / sm;
  }
  __syncthreads();
  for (int e = tid; e < E_; e += 256) {
    const float* mb = memory + (size_t)b * S_ * E_ + e;
    float acc = 0.f;
    for (int s = 0; s < S_; ++s) acc += en[s] * mb[(size_t)s * E_];
    context[(size_t)bt * E_ + e] = acc * ssum;
  }
}

// h_n = stack(h1_T, h2_T), c_n = stack(c0, c1)
__global__ void finalize_kernel(const float* __restrict__ y1,
                                const float* __restrict__ y2,
                                const float* __restrict__ c0,
                                const float* __restrict__ c1,
                                float* __restrict__ hn, float* __restrict__ cn) {
  int idx = blockIdx.x * blockDim.x + threadIdx.x;  // 0 .. B*H-1
  int b = idx >> 9, h = idx & (H_ - 1);
  hn[idx] = y1[((size_t)b * T_ + (T_ - 1)) * H_ + h];
  hn[B_ * H_ + idx] = y2[((size_t)b * T_ + (T_ - 1)) * H_ + h];
  cn[idx] = c0[idx];
  cn[B_ * H_ + idx] = c1[idx];
}

// ---------------------------------------------------------------------------
extern "C" void kernel_launch(void* const* d_in, const int* in_sizes, int n_in,
                              void* d_out, int out_size, void* d_ws,
                              size_t ws_size, hipStream_t stream) {
  const int* tgt = (const int*)d_in[0];
  const float* memory = (const float*)d_in[1];
  const unsigned char* mask = (const unsigned char*)d_in[2];
  const float* emb = (const float*)d_in[3];
  const float* Wih0 = (const float*)d_in[4];
  const float* Whh0 = (const float*)d_in[5];
  const float* bih0 = (const float*)d_in[6];
  const float* bhh0 = (const float*)d_in[7];
  const float* Wih1 = (const float*)d_in[8];
  const float* Whh1 = (const float*)d_in[9];
  const float* bih1 = (const float*)d_in[10];
  const float* bhh1 = (const float*)d_in[11];
  const float* Wh = (const float*)d_in[12];
  const float* bh = (const float*)d_in[13];
  const float* Ws = (const float*)d_in[14];
  const float* bs = (const float*)d_in[15];
  const float* v = (const float*)d_in[16];
  const float* vb = (const float*)d_in[17];
  const float* Wout = (const float*)d_in[18];
  const float* bout = (const float*)d_in[19];

  float* ws = (float*)d_ws;
  size_t off = 0;
  float* x = ws + off;      off += (size_t)B_ * T_ * E_;
  float* xg0 = ws + off;    off += (size_t)B_ * T_ * 4 * H_;
  float* hinit = ws + off;  off += (size_t)B_ * H_;
  float* c0buf = ws + off;  off += (size_t)B_ * H_;
  float* c1buf = ws + off;  off += (size_t)B_ * H_;
  float* y1 = ws + off;     off += (size_t)B_ * T_ * H_;
  float* y2 = ws + off;     off += (size_t)B_ * T_ * H_;
  float* encf = ws + off;   off += (size_t)B_ * S_ * H_;
  float* decf = ws + off;   off += (size_t)B_ * T_ * H_;
  float* ctx = ws + off;    off += (size_t)B_ * T_ * E_;

  float* logits = (float*)d_out;              // [B,T,V]
  float* hn = logits + (size_t)B_ * T_ * V_;  // [2,B,H]
  float* cn = hn + 2 * (size_t)B_ * H_;       // [2,B,H]

  embed_kernel<<<B_ * T_, E_ / 4, 0, stream>>>(tgt, emb, x);
  init_kernel<<<(B_ * H_) / 256, 256, 0, stream>>>(memory, hinit, c0buf, c1buf);

  // Hoisted layer-0 input GEMM: xg0 = x @ Wih0^T + (bih0+bhh0), [B*T, 4H]
  gemm_wt_kernel<<<dim3((B_ * T_) / TM, (4 * H_) / TN), 128, GEMM_LDS_BYTES,
                   stream>>>(x, E_, E_, nullptr, 0, 0, Wih0, E_, bih0, bhh0,
                             xg0, 4 * H_);

  // Recurrent scans. h history lives in y1/y2 (rows strided T*H between
  // batches); c buffers update in place (slice-private per workgroup).
  for (int t = 0; t < T_; ++t) {
    const float* hp = (t == 0) ? hinit : (y1 + (size_t)(t - 1) * H_);
    long hs = (t == 0) ? (long)H_ : (long)T_ * H_;
    lstm_step0_kernel<<<H_ / 16, 32, 0, stream>>>(
        xg0 + (size_t)t * 4 * H_, (long)T_ * 4 * H_, hp, hs, Whh0, c0buf,
        y1 + (size_t)t * H_, (long)T_ * H_);
  }
  for (int t = 0; t < T_; ++t) {
    const float* hp = (t == 0) ? hinit : (y2 + (size_t)(t - 1) * H_);
    long hs = (t == 0) ? (long)H_ : (long)T_ * H_;
    lstm_step1_kernel<<<H_ / 16, 32, 0, stream>>>(
        y1 + (size_t)t * H_, (long)T_ * H_, Wih1, hp, hs, Whh1, bih1, bhh1,
        c1buf, y2 + (size_t)t * H_, (long)T_ * H_);
  }

  // Attention features.
  gemm_wt_kernel<<<dim3((B_ * S_) / TM, H_ / TN), 128, GEMM_LDS_BYTES,
                   stream>>>(memory, E_, E_, nullptr, 0, 0, Wh, E_, bh, nullptr,
                             encf, H_);
  gemm_wt_kernel<<<dim3((B_ * T_) / TM, H_ / TN), 128, GEMM_LDS_BYTES,
                   stream>>>(y2, H_, H_, nullptr, 0, 0, Ws, H_, bs, nullptr,
                             decf, H_);

  attn_kernel<<<B_ * T_, 256, 0, stream>>>(encf, decf, memory, mask, v, vb, ctx);

  // logits = [y2 | context] @ Wout^T + bout (split-K concat, no materialize).
  gemm_wt_kernel<<<dim3((B_ * T_) / TM, V_ / TN), 128, GEMM_LDS_BYTES,
                   stream>>>(y2, H_, H_, ctx, E_, E_, Wout, H_ + E_, bout,
                             nullptr, logits, V_);

  finalize_kernel<<<(B_ * H_) / 256, 256, 0, stream>>>(y1, y2, c0buf, c1buf,
                                                       hn, cn);
}